// Logistic_70677981823177
// MI455X (gfx1250) — compile-verified
//
#include <hip/hip_runtime.h>

// Problem constants (from reference): x:(512,784) f32, W:(784*784,10) f32, b:(10,)
#define D    784
#define C    10
#define NB   512
#define DD   (D * D)        // 614656
#define XS   788            // padded LDS row stride (dwords): 788%64=20 -> conflict-free b64 A-loads

typedef float v2f __attribute__((ext_vector_type(2)));
typedef float v8f __attribute__((ext_vector_type(8)));

// ---------------------------------------------------------------------------
// Transpose W[(i*D+j)*C+k] -> Wt[k*DD + j*D + i]  (makes GEMM B-loads coalesced)
// LDS-tiled: reads are 40B-contiguous per thread (all 10 k at once), writes are
// 128B-coalesced runs over i.
// ---------------------------------------------------------------------------
#define TI 32
#define TJ 32
__global__ __launch_bounds__(256) void transpose_W(const float* __restrict__ W,
                                                   float* __restrict__ Wt) {
    __shared__ float t[C][TJ][TI + 1];          // 10*32*33*4 = 42240 B
    const int i0 = blockIdx.x * TI;
    const int j0 = blockIdx.y * TJ;

    // Load: consecutive threads -> consecutive j -> contiguous 40B chunks.
    for (int p = threadIdx.x; p < TI * TJ; p += 256) {
        const int ti = p / TJ;
        const int tj = p % TJ;
        const int i = i0 + ti, j = j0 + tj;
        if (i < D && j < D) {
            const float* src = W + ((size_t)i * D + j) * C;
            #pragma unroll
            for (int k = 0; k < C; ++k) t[k][tj][ti] = src[k];
        }
    }
    __syncthreads();

    // Store: consecutive threads -> consecutive i -> 128B coalesced lines.
    for (int q = threadIdx.x; q < C * TJ * TI; q += 256) {
        const int ti = q % TI;
        const int tj = (q / TI) % TJ;
        const int k  = q / (TI * TJ);
        const int i = i0 + ti, j = j0 + tj;
        if (i < D && j < D)
            Wt[(size_t)k * DD + (size_t)j * D + i] = t[k][tj][ti];
    }
}

// ---------------------------------------------------------------------------
// Fused bilinear kernel: block = (batch-tile of 16, output class k).
//   Phase 1 per column tile (16 i-values, fixed k):
//       Dtile[m][n] = sum_j x[b0+m, j] * W[i0+n, j, k]     via V_WMMA_F32_16X16X4_F32
//   Phase 2: y[b0+m, k] += sum_n Dtile[m][n] * x[b0+m, i0+n]  (shuffle reduction)
// 784 = 49 tiles of 16 columns; 8 waves split the 49 tiles; 196 WMMA per tile.
// Deterministic: no atomics, every y element written exactly once.
// ---------------------------------------------------------------------------
template <bool TRANSPOSED>
__global__ __launch_bounds__(256) void bilinear_wmma(const float* __restrict__ x,
                                                     const float* __restrict__ Wsrc,
                                                     const float* __restrict__ bias,
                                                     float* __restrict__ y) {
    __shared__ float xs[16 * XS];               // 50432 B x-tile, padded stride
    __shared__ float part[8][16];               // per-wave partial y rows

    const int btile = blockIdx.x;               // 0..31
    const int k     = blockIdx.y;               // 0..9
    const int b0    = btile * 16;
    const int tid   = threadIdx.x;
    const int wave  = tid >> 5;                 // 0..7
    const int lane  = tid & 31;
    const int half  = lane >> 4;                // 0 | 1
    const int nl    = lane & 15;                // n (col) / m (row) within fragment

    // --- stage x tile [16][784] into LDS (rows are contiguous in global) ---
    {
        const int m  = tid >> 4;                // 16 threads per row
        const int c0 = tid & 15;
        const float* row = x + (size_t)(b0 + m) * D;
        for (int c = c0; c < D; c += 16)        // coalesced 64B per 16-group
            xs[m * XS + c] = row[c];
    }
    __syncthreads();

    float acc[8];
    #pragma unroll
    for (int r = 0; r < 8; ++r) acc[r] = 0.0f;

    for (int iblk = wave; iblk < 49; iblk += 8) {
        const int i0 = iblk * 16;
        v8f c = {};                             // f32 accumulator tile

        // K-loop over j: 196 steps of K=4
        #pragma unroll 4
        for (int j0 = 0; j0 < D; j0 += 4) {
            const int jj = j0 + 2 * half;       // lanes 0-15: K=0,1  lanes 16-31: K=2,3
            // A fragment: x[b0+nl, jj..jj+1]  (ds_load_b64, bank-conflict-free)
            v2f a = *(const v2f*)&xs[nl * XS + jj];
            // B fragment: W[i0+nl, jj..jj+1, k]
            v2f bf;
            if (TRANSPOSED) {
                const float* wp = Wsrc + (size_t)k * DD + (size_t)jj * D + (i0 + nl);
                bf.x = wp[0];                   // coalesced over nl
                bf.y = wp[D];
            } else {
                const float* wp = Wsrc + ((size_t)(i0 + nl) * D + jj) * C + k;
                bf.x = wp[0];                   // gather fallback (L2-resident W)
                bf.y = wp[C];
            }
            c = __builtin_amdgcn_wmma_f32_16x16x4_f32(
                    false, a, false, bf, (short)0, c, false, false);
        }

        // epilogue: weight column n by x[b0+m, i0+n], reduce over the 16 lanes
        #pragma unroll
        for (int r = 0; r < 8; ++r) {
            const int m = r + 8 * half;         // C/D layout: lanes>=16 hold M=r+8
            float p = c[r] * xs[m * XS + i0 + nl];
            p += __shfl_xor(p, 1, 32);
            p += __shfl_xor(p, 2, 32);
            p += __shfl_xor(p, 4, 32);
            p += __shfl_xor(p, 8, 32);          // stays within each 16-lane half
            acc[r] += p;
        }
    }

    // per-wave partials -> LDS (lane 0: rows 0..7, lane 16: rows 8..15)
    if (nl == 0) {
        #pragma unroll
        for (int r = 0; r < 8; ++r) part[wave][r + 8 * half] = acc[r];
    }
    __syncthreads();

    // fixed-order cross-wave reduction + bias; single writer per output element
    if (tid < 16) {
        float s = bias[k];
        #pragma unroll
        for (int w = 0; w < 8; ++w) s += part[w][tid];
        y[(size_t)(b0 + tid) * C + k] = s;
    }
}

// ---------------------------------------------------------------------------
extern "C" void kernel_launch(void* const* d_in, const int* in_sizes, int n_in,
                              void* d_out, int out_size, void* d_ws, size_t ws_size,
                              hipStream_t stream) {
    (void)in_sizes; (void)n_in; (void)out_size;
    const float* x    = (const float*)d_in[0];
    const float* W    = (const float*)d_in[1];
    const float* bias = (const float*)d_in[2];
    float*       y    = (float*)d_out;

    const size_t wt_bytes = (size_t)DD * C * sizeof(float);   // 24.6 MB
    dim3 grid(NB / 16, C);        // 32 x 10 blocks
    dim3 block(256);              // 8 waves (wave32)

    if (ws_size >= wt_bytes) {
        float* Wt = (float*)d_ws;
        dim3 tgrid((D + TI - 1) / TI, (D + TJ - 1) / TJ);     // 25 x 25
        transpose_W<<<tgrid, 256, 0, stream>>>(W, Wt);
        bilinear_wmma<true><<<grid, block, 0, stream>>>(x, Wt, bias, y);
    } else {
        bilinear_wmma<false><<<grid, block, 0, stream>>>(x, W, bias, y);
    }
}